// _Adder2d_41729902248401
// MI455X (gfx1250) — compile-verified
//
#include <hip/hip_runtime.h>

// AdderNet 2D (L1-distance "convolution"), fp32, N=4 Ci=64 H=W=28 Co=128 K=3 pad=1.
// out[n,co,i,j] = -sum_{ci,kh,kw} |x_pad[n,ci,i+kh,j+kw] - w[co,ci,kh,kw]|
//
// VALU-bound (~462M f32 ops, 2.7MB working set, fully L2-resident).
// WMMA is inapplicable (L1 distance is not bilinear); the CDNA5 feature used
// is the async global->LDS copy path (ASYNCcnt + s_wait_asynccnt).
//
// Blocking: block = (n, 8 output channels, 14-row spatial half).
//   grid = 4 * 16 * 2 = 128 blocks x 256 threads (8 wave32 / block).
//   LDS: 8 padded x-planes 16x30 (15.4KB) + 8x8x9 weights (2.3KB).
//   Per ci: 288 VALU (8co x 9taps x 2 x 2px) vs ~30 LDS reads -> VALU-bound.

#define N_      4
#define CI_     64
#define HW_     28
#define CO_     128
#define PP_     784        // 28*28 pixels per (n,co) plane
#define HROWS_  14         // output rows per block (half plane)
#define PX_     392        // 14*28 pixels per block
#define PLX_    480        // 16*30 padded LDS plane
#define CC_     8          // ci chunk staged in LDS
#define COB_    8          // output channels per block
#define NP_     2          // pixels per thread (256*2 >= 392)
#define THREADS_ 256

// ---- async global->LDS helpers (gfx1250) --------------------------------
#if defined(__has_builtin)
#if __has_builtin(__builtin_amdgcn_global_load_async_to_lds_b32)
#define HAVE_ASYNC_LDS 1
#endif
#endif

typedef __attribute__((address_space(1))) int* as1ip;   // global int*
typedef __attribute__((address_space(3))) int* as3ip;   // LDS int*

__device__ __forceinline__ void lds_fill_b32(const float* g, float* l) {
#if HAVE_ASYNC_LDS
  // per-lane: copies 4 bytes from global address g to LDS address l  (tracked by ASYNCcnt, no VGPR round-trip)
  __builtin_amdgcn_global_load_async_to_lds_b32(
      (as1ip)(unsigned long long)(const void*)g,
      (as3ip)(unsigned)(unsigned long long)(void*)l,
      0, 0);
#else
  *l = *g;
#endif
}

__device__ __forceinline__ void async_wait0() {
#if HAVE_ASYNC_LDS
#if defined(__has_builtin)
#if __has_builtin(__builtin_amdgcn_s_wait_asynccnt)
  __builtin_amdgcn_s_wait_asynccnt(0);
#else
  asm volatile("s_wait_asynccnt 0" ::: "memory");
#endif
#else
  asm volatile("s_wait_asynccnt 0" ::: "memory");
#endif
#endif
}

// -------------------------------------------------------------------------
__global__ __launch_bounds__(THREADS_) void adder2d_kernel(
    const float* __restrict__ x,      // [4,64,28,28]
    const float* __restrict__ w,      // [128,64,3,3]
    float* __restrict__ out) {        // [4,128,28,28]
  __shared__ float xs[CC_ * PLX_ + 4];       // 8 padded 16x30 planes (+pad)
  __shared__ float ws[COB_ * CC_ * 9];       // 8co x 8ci x 3x3

  const int tid  = threadIdx.x;
  const int b    = blockIdx.x;
  const int half = b & 1;                    // 0: rows 0..13, 1: rows 14..27
  const int cob  = (b >> 1) & 15;
  const int n    = b >> 5;
  const int co0  = cob * COB_;
  const int h0   = half * HROWS_;            // first output row of this block
  const int rlo  = (half == 0) ? 1 : 0;      // first valid padded-local LDS row

  // Zero the padded planes once; chunk loads only rewrite the 15 valid
  // interior rows, so the halo (incl. the off-image row) stays zero.
  for (int e = tid; e < CC_ * PLX_; e += THREADS_) xs[e] = 0.0f;

  int li[NP_], lj[NP_];
  float acc[NP_][COB_];
#pragma unroll
  for (int u = 0; u < NP_; ++u) {
    const int p = tid + u * THREADS_;
    li[u] = p / HW_;                         // local output row 0..13
    lj[u] = p - li[u] * HW_;                 // output col 0..27
#pragma unroll
    for (int c = 0; c < COB_; ++c) acc[u][c] = 0.0f;
  }

  for (int cc = 0; cc < CI_ / CC_; ++cc) {
    const int ci0 = cc * CC_;
    __syncthreads();  // prior chunk compute done (first iter: zero-fill done)

    // Stage x: per plane, the 15 global rows that exist for this half.
    // LDS padded-local row r <-> padded row h0+r <-> global row h0+r-1.
    for (int e = tid; e < CC_ * 15 * HW_; e += THREADS_) {
      const int ci  = e / (15 * HW_);
      const int rem = e - ci * (15 * HW_);
      const int rr  = rem / HW_;             // 0..14
      const int c   = rem - rr * HW_;        // 0..27
      const int r   = rr + rlo;              // padded-local LDS row
      const int gr  = h0 - 1 + r;            // global row, in [0,27] by constr.
      lds_fill_b32(x + ((size_t)(n * CI_ + ci0 + ci) * HW_ + gr) * HW_ + c,
                   &xs[ci * PLX_ + r * 30 + (c + 1)]);
    }
    // Stage weights w[co0..co0+7, ci0..ci0+7, :, :] (576 elems).
    for (int e = tid; e < COB_ * CC_ * 9; e += THREADS_) {
      const int c  = e / (CC_ * 9);
      const int rr = e - c * (CC_ * 9);
      const int ci = rr / 9;
      const int k  = rr - ci * 9;
      lds_fill_b32(w + (((size_t)(co0 + c) * CI_ + ci0 + ci) * 9 + k), &ws[e]);
    }
    async_wait0();
    __syncthreads();

    for (int ci = 0; ci < CC_; ++ci) {
      float wr[COB_][9];
#pragma unroll
      for (int c = 0; c < COB_; ++c)
#pragma unroll
        for (int k = 0; k < 9; ++k) wr[c][k] = ws[(c * CC_ + ci) * 9 + k];

#pragma unroll
      for (int u = 0; u < NP_; ++u) {
        if (tid + u * THREADS_ < PX_) {
          const float* bse = &xs[ci * PLX_ + li[u] * 30 + lj[u]];
          float xv[9];
#pragma unroll
          for (int kh = 0; kh < 3; ++kh)
#pragma unroll
            for (int kw = 0; kw < 3; ++kw) xv[kh * 3 + kw] = bse[kh * 30 + kw];
#pragma unroll
          for (int c = 0; c < COB_; ++c)
#pragma unroll
            for (int k = 0; k < 9; ++k)
              acc[u][c] += __builtin_fabsf(xv[k] - wr[c][k]);
        }
      }
    }
  }

#pragma unroll
  for (int u = 0; u < NP_; ++u) {
    const int p = tid + u * THREADS_;
    if (p < PX_) {
      const int gp = h0 * HW_ + p;           // global pixel index in plane
#pragma unroll
      for (int c = 0; c < COB_; ++c)
        out[(size_t)(n * CO_ + co0 + c) * PP_ + gp] = -acc[u][c];
    }
  }
}

extern "C" void kernel_launch(void* const* d_in, const int* in_sizes, int n_in,
                              void* d_out, int out_size, void* d_ws, size_t ws_size,
                              hipStream_t stream) {
  (void)in_sizes; (void)n_in; (void)out_size; (void)d_ws; (void)ws_size;
  const float* x = (const float*)d_in[0];   // [4,64,28,28]
  const float* w = (const float*)d_in[1];   // [128,64,3,3]
  float* out = (float*)d_out;               // [4,128,28,28]
  dim3 grid(N_ * (CO_ / COB_) * 2);         // 128 blocks
  adder2d_kernel<<<grid, THREADS_, 0, stream>>>(x, w, out);
}